// GATConvwithEdgeFeat_9818295238978
// MI455X (gfx1250) — compile-verified
//
#include <hip/hip_runtime.h>
#include <hip/hip_bf16.h>

typedef __attribute__((ext_vector_type(2))) float v2f;
typedef __attribute__((ext_vector_type(8))) float v8f;

#define FIN 128
#define HD 128
#define KCAT 256
#define NEG_SLOPE 0.2f
#define ETILES 4           // edge row-tiles per block (64 edges/block)

// ---- CDNA5 float atomics (performed at L2; DEV scope) ----
__device__ __forceinline__ void atomic_fmax_dev(float* addr, float v) {
  asm volatile("global_atomic_max_num_f32 %0, %1, off scope:SCOPE_DEV"
               :: "v"(addr), "v"(v) : "memory");
}
__device__ __forceinline__ void atomic_fadd_dev(float* addr, float v) {
  asm volatile("global_atomic_add_f32 %0, %1, off scope:SCOPE_DEV"
               :: "v"(addr), "v"(v) : "memory");
}

// ---- 1) init: mx=-inf, denom=0, out=0 ----
__global__ void gat_init_kernel(float* __restrict__ mx, float* __restrict__ denom,
                                float* __restrict__ out, int n) {
  int i = blockIdx.x * blockDim.x + threadIdx.x;
  if (i < n) {
    mx[i] = -__builtin_inff();
    denom[i] = 0.0f;
    out[i] = 0.0f;
  }
}

// ---- 2) nh = dst_feat @ W_dst  (fp32 WMMA, 16x16 tile per wave) ----
__global__ __launch_bounds__(256)
void gat_nh_gemm_kernel(const float* __restrict__ dst_feat,
                        const float* __restrict__ W_dst,
                        float* __restrict__ nh, int N) {
  __shared__ float Xs[16][FIN + 4];   // stride 132 (132%64==4 -> conflict-free v2f reads)
  const int tid = threadIdx.x;
  const int n0 = blockIdx.x * 16;
  {
    int r = tid >> 4;
    int c4 = (tid & 15) << 2;
    int row = n0 + r; if (row >= N) row = N - 1;
    const float* p = dst_feat + (size_t)row * FIN;
    *(float4*)&Xs[r][c4]      = *(const float4*)(p + c4);
    *(float4*)&Xs[r][c4 + 64] = *(const float4*)(p + c4 + 64);
  }
  __syncthreads();
  const int lane = tid & 31, wave = tid >> 5;
  const int half = lane >> 4, mrow = lane & 15;
  const int ncol = wave * 16 + mrow;
  v8f acc = {};
  #pragma unroll 4
  for (int kb = 0; kb < FIN; kb += 4) {
    int ka = kb + half * 2;
    v2f a = *(const v2f*)&Xs[mrow][ka];
    v2f b;
    b.x = W_dst[(size_t)ka * HD + ncol];
    b.y = W_dst[(size_t)(ka + 1) * HD + ncol];
    acc = __builtin_amdgcn_wmma_f32_16x16x4_f32(false, a, false, b, (short)0, acc,
                                                false, false);
  }
  #pragma unroll
  for (int r = 0; r < 8; ++r) {
    int row = n0 + r + half * 8;
    if (row < N) nh[(size_t)row * HD + ncol] = acc[r];
  }
}

// ---- 3) m = concat(src[src_idx], edge) @ W_src  (fp32 WMMA, 4 tiles/wave)
//         fused: score = leaky_relu(m + nh[dst]) ; atomic max into mx ----
__global__ __launch_bounds__(256)
void gat_edge_gemm_kernel(const float* __restrict__ src_feat,
                          const float* __restrict__ edge_feat,
                          const float* __restrict__ W_src,
                          const int* __restrict__ src_idx,
                          const int* __restrict__ dst_idx,
                          const float* __restrict__ nh,
                          float* __restrict__ m_out,
                          float* __restrict__ mx, int E) {
  __shared__ float Xs[16 * ETILES][KCAT + 4];  // 64 x 260 floats = 66.5 KB
  const int tid = threadIdx.x;
  const int e0 = blockIdx.x * (16 * ETILES);
  // stage 64 concatenated rows into LDS (coalesced float4 per row segment)
  {
    int r = tid >> 4;
    int c4 = (tid & 15) << 2;
    #pragma unroll
    for (int g = 0; g < ETILES; ++g) {
      int row = g * 16 + r;
      int e = e0 + row; if (e >= E) e = E - 1;
      const float* ps = src_feat + (size_t)src_idx[e] * FIN;
      const float* pe = edge_feat + (size_t)e * FIN;
      *(float4*)&Xs[row][c4]       = *(const float4*)(ps + c4);
      *(float4*)&Xs[row][c4 + 64]  = *(const float4*)(ps + c4 + 64);
      *(float4*)&Xs[row][c4 + 128] = *(const float4*)(pe + c4);
      *(float4*)&Xs[row][c4 + 192] = *(const float4*)(pe + c4 + 64);
    }
  }
  __syncthreads();
  const int lane = tid & 31, wave = tid >> 5;
  const int half = lane >> 4, mrow = lane & 15;
  const int ncol = wave * 16 + mrow;
  const float* Wc = W_src + ncol;

  v8f acc[ETILES] = {};
  #pragma unroll 2
  for (int kb = 0; kb < KCAT; kb += 4) {
    int ka = kb + half * 2;
    v2f b;                                  // one B k-slice reused by 4 A tiles
    b.x = Wc[(size_t)ka * HD];
    b.y = Wc[(size_t)(ka + 1) * HD];
    #pragma unroll
    for (int g = 0; g < ETILES; ++g) {
      v2f a = *(const v2f*)&Xs[g * 16 + mrow][ka];
      acc[g] = __builtin_amdgcn_wmma_f32_16x16x4_f32(false, a, false, b, (short)0,
                                                     acc[g], false, false);
    }
  }

  // fused epilogue: store m, compute leaky score, atomic-max into mx
  if (e0 + 16 * ETILES <= E) {            // uniform fast path (no per-row guards)
    #pragma unroll
    for (int g = 0; g < ETILES; ++g) {
      #pragma unroll
      for (int r = 0; r < 8; ++r) {
        int e = e0 + g * 16 + r + half * 8;
        float mval = acc[g][r];
        m_out[(size_t)e * HD + ncol] = mval;
        int d = dst_idx[e];
        float s = mval + nh[(size_t)d * HD + ncol];
        s = (s > 0.0f) ? s : NEG_SLOPE * s;
        atomic_fmax_dev(&mx[(size_t)d * HD + ncol], s);
      }
    }
  } else {                                 // ragged tail block
    #pragma unroll
    for (int g = 0; g < ETILES; ++g) {
      #pragma unroll
      for (int r = 0; r < 8; ++r) {
        int e = e0 + g * 16 + r + half * 8;
        if (e < E) {
          float mval = acc[g][r];
          m_out[(size_t)e * HD + ncol] = mval;
          int d = dst_idx[e];
          float s = mval + nh[(size_t)d * HD + ncol];
          s = (s > 0.0f) ? s : NEG_SLOPE * s;
          atomic_fmax_dev(&mx[(size_t)d * HD + ncol], s);
        }
      }
    }
  }
}

// ---- 4) ex = exp(score - mx[dst]); denom += ex; out += m*ex  (float4/thread) ----
__global__ __launch_bounds__(256)
void gat_exp_accum_kernel(const float* __restrict__ m,
                          const int* __restrict__ dst_idx,
                          const float* __restrict__ nh,
                          const float* __restrict__ mx,
                          float* __restrict__ denom,
                          float* __restrict__ out, int E) {
  long long idx = (long long)blockIdx.x * blockDim.x + threadIdx.x;
  int e = (int)(idx >> 5);                 // 32 threads x 4 channels = 128 per edge
  int c = (int)(idx & 31) << 2;
  if (e >= E) return;
  int d = dst_idx[e];
  const float4 mv = *(const float4*)(m  + (size_t)e * HD + c);
  const float4 nv = *(const float4*)(nh + (size_t)d * HD + c);
  const float4 xv = *(const float4*)(mx + (size_t)d * HD + c);
  float* dp = denom + (size_t)d * HD + c;
  float* op = out   + (size_t)d * HD + c;
  #pragma unroll
  for (int j = 0; j < 4; ++j) {
    float mval = (&mv.x)[j];
    float s = mval + (&nv.x)[j];
    s = (s > 0.0f) ? s : NEG_SLOPE * s;
    float ex = __expf(s - (&xv.x)[j]);
    atomic_fadd_dev(dp + j, ex);
    atomic_fadd_dev(op + j, mval * ex);
  }
}

// ---- 5) out = out/denom + bias (empty segment -> bias) ----
__global__ void gat_finalize_kernel(const float* __restrict__ denom,
                                    const float* __restrict__ bias,
                                    float* __restrict__ out, int n) {
  int i = blockIdx.x * blockDim.x + threadIdx.x;
  if (i < n) {
    float dn = denom[i];
    float o  = out[i];
    float b  = bias[i & (HD - 1)];
    out[i] = (dn > 0.0f) ? (o / dn + b) : b;
  }
}

extern "C" void kernel_launch(void* const* d_in, const int* in_sizes, int n_in,
                              void* d_out, int out_size, void* d_ws, size_t ws_size,
                              hipStream_t stream) {
  const float* src_feat  = (const float*)d_in[0];
  const float* dst_feat  = (const float*)d_in[1];
  const float* edge_feat = (const float*)d_in[2];
  const float* W_src     = (const float*)d_in[3];
  const float* W_dst     = (const float*)d_in[4];
  const float* bias      = (const float*)d_in[5];
  const int*   src_idx   = (const int*)d_in[6];
  const int*   dst_idx   = (const int*)d_in[7];
  float* out = (float*)d_out;

  const int E = in_sizes[6];
  const int N = in_sizes[1] / FIN;
  const int nND = N * HD;

  // workspace layout (floats): nh | mx | denom | m
  float* ws    = (float*)d_ws;
  float* nh    = ws;
  float* mx    = nh + (size_t)nND;
  float* denom = mx + (size_t)nND;
  float* m     = denom + (size_t)nND;

  hipLaunchKernelGGL(gat_init_kernel, dim3((nND + 255) / 256), dim3(256), 0, stream,
                     mx, denom, out, nND);
  hipLaunchKernelGGL(gat_nh_gemm_kernel, dim3((N + 15) / 16), dim3(256), 0, stream,
                     dst_feat, W_dst, nh, N);
  const int EB = 16 * ETILES;
  hipLaunchKernelGGL(gat_edge_gemm_kernel, dim3((E + EB - 1) / EB), dim3(256), 0, stream,
                     src_feat, edge_feat, W_src, src_idx, dst_idx, nh, m, mx, E);
  long long tot = (long long)E * 32;       // 32 threads per edge (4 ch each)
  hipLaunchKernelGGL(gat_exp_accum_kernel, dim3((unsigned)((tot + 255) / 256)), dim3(256), 0, stream,
                     m, dst_idx, nh, mx, denom, out, E);
  hipLaunchKernelGGL(gat_finalize_kernel, dim3((nND + 255) / 256), dim3(256), 0, stream,
                     denom, bias, out, nND);
}